// GeometryAwareAssign_50165218017444
// MI455X (gfx1250) — compile-verified
//
#include <hip/hip_runtime.h>
#include <hip/hip_bf16.h>
#include <stdint.h>

// Problem constants (match setup_inputs)
#define K_B 8
#define K_P 4096
#define K_T 64
#define K_N 72
#define K_NF 78   // 6 + N
#define K_Q 10

typedef float v2f __attribute__((ext_vector_type(2)));
typedef float v8f __attribute__((ext_vector_type(8)));

// LDS generic pointers on gfx1250: low 32 bits of the generic address are the
// LDS byte offset (aperture in the high bits) -- see ISA 10.2 aperture rules.
__device__ __forceinline__ uint32_t lds_off32(const void* p) {
  return (uint32_t)(uintptr_t)p;
}

// CDNA5 async global->LDS copy (tracked with ASYNCcnt). GVS addressing:
// saddr (SGPR pair) + 32-bit per-lane offset; vdst VGPR carries the LDS byte
// address. 16-byte granules, 16B-aligned on both sides.
__device__ __forceinline__ void async_copy_b128(uint32_t lds_byte_off,
                                                uint32_t g_byte_off,
                                                const void* gbase) {
  asm volatile("global_load_async_to_lds_b128 %0, %1, %2"
               :: "v"(lds_byte_off), "v"(g_byte_off), "s"(gbase)
               : "memory");
}
__device__ __forceinline__ void async_wait0() {
  asm volatile("s_wait_asynccnt 0" ::: "memory");
}

// ---------------------------------------------------------------------------
// Kernel A: fused pairwise cost + line-IoU.
// grid = (P/16, B), block = 128 (4 waves). Block handles 16 priors x 64 GTs;
// wave w handles GT tile [16w, 16w+16). Lane -> (g = lane%16 GT, 8 prior rows
// 8*(lane/16)+j), matching the WMMA f32 16x16 C/D VGPR layout.
// ---------------------------------------------------------------------------
struct __align__(16) SmemA {
  float gt[K_T * K_NF];   // 4992 floats, 19968 B (16B-aligned chunks)
  float pr[16 * K_NF];    // 1248 floats, 4992 B
  float cls[16];
  float pp[16];
  float maskf[K_T];
};

__global__ __launch_bounds__(128) void cost_iou_kernel(
    const float* __restrict__ preds, const float* __restrict__ targets,
    const int* __restrict__ masks, const int* __restrict__ imgw_p,
    float* __restrict__ cost, float* __restrict__ iou_out) {
  __shared__ SmemA sm;
  const int b = blockIdx.y;
  const int p0 = blockIdx.x * 16;
  const int tid = threadIdx.x;
  const float W = (float)(*imgw_p);
  const float Wm1 = W - 1.0f;

  const float* gsrc = targets + (size_t)b * K_T * K_NF;       // 16B aligned
  const float* psrc = preds + ((size_t)b * K_P + p0) * K_NF;  // 16B aligned
  const uint32_t gt_lds = lds_off32(sm.gt);
  const uint32_t pr_lds = lds_off32(sm.pr);
  for (int c = tid; c < (K_T * K_NF) / 4; c += 128)   // 1248 x b128
    async_copy_b128(gt_lds + c * 16, c * 16, gsrc);
  for (int c = tid; c < (16 * K_NF) / 4; c += 128)    // 312 x b128
    async_copy_b128(pr_lds + c * 16, c * 16, psrc);
  async_wait0();
  __syncthreads();

  // Per-prior scalars (softmax cls cost, |xy|^2)
  if (tid < 16) {
    const float l0 = sm.pr[tid * K_NF + 0];
    const float l1 = sm.pr[tid * K_NF + 1];
    const float m = fmaxf(l0, l1);
    const float e0 = __expf(l0 - m), e1 = __expf(l1 - m);
    const float s = e1 / (e0 + e1);
    sm.cls[tid] = -__logf(fmaxf(s, 1e-8f));
    const float px = sm.pr[tid * K_NF + 3];
    const float py = sm.pr[tid * K_NF + 2];
    sm.pp[tid] = px * px + py * py;
  }
  if (tid < K_T) sm.maskf[tid] = (float)masks[b * K_T + tid];
  __syncthreads();

  const int lane = tid & 31;
  const int g0 = (tid >> 5) * 16;
  const int g = g0 + (lane & 15);        // this lane's GT column
  const int prow0 = (lane >> 4) * 8;     // this lane's first prior row

  // ---- WMMA: cross term of pairwise distance, D = A(16x4) x B(4x16) ----
  // A lanes 0-15: (K0,K1) = (px,py) of row M=lane; lanes 16-31: K2,K3 = 0.
  v2f amat = {0.f, 0.f}, bmat = {0.f, 0.f};
  if (lane < 16) {
    amat.x = sm.pr[lane * K_NF + 3];
    amat.y = sm.pr[lane * K_NF + 2];
    const int gg = g0 + lane;
    bmat.x = sm.gt[gg * K_NF + 3];
    bmat.y = sm.gt[gg * K_NF + 2];
  }
  float crossv[8];
#if __has_builtin(__builtin_amdgcn_wmma_f32_16x16x4_f32)
  {
    v8f acc = {0.f, 0.f, 0.f, 0.f, 0.f, 0.f, 0.f, 0.f};
    acc = __builtin_amdgcn_wmma_f32_16x16x4_f32(
        false, amat, false, bmat, (short)0, acc, false, false);
#pragma unroll
    for (int j = 0; j < 8; ++j) crossv[j] = acc[j];
  }
#else
  {
    const float fx = sm.gt[g * K_NF + 3], fy = sm.gt[g * K_NF + 2];
#pragma unroll
    for (int j = 0; j < 8; ++j) {
      const int pl = prow0 + j;
      crossv[j] = sm.pr[pl * K_NF + 3] * fx + sm.pr[pl * K_NF + 2] * fy;
    }
  }
#endif

  // ---- line IoU: S = sum_valid |p - t|, V = #valid (branchless, b64 LDS) --
  // Index row*78 + 6 + n is even for even n (78, 6, prow0 all even), so
  // float2 (ds_load_b64) access at n stepping by 2 is 8B-aligned.
  const int gbase = g * K_NF + 6;
  const int pbase = prow0 * K_NF + 6;
  float S[8];
#pragma unroll
  for (int j = 0; j < 8; ++j) S[j] = 0.f;
  float VC = 0.f;
  for (int n = 0; n < K_N; n += 2) {
    const float2 t2 = *reinterpret_cast<const float2*>(&sm.gt[gbase + n]);
    const float t0 = t2.x * Wm1;
    const float t1 = t2.y * Wm1;
    const float w0 = (t0 >= 0.f && t0 < W) ? 1.f : 0.f;
    const float w1 = (t1 >= 0.f && t1 < W) ? 1.f : 0.f;
    VC += w0 + w1;
#pragma unroll
    for (int j = 0; j < 8; ++j) {
      const float2 p2 =
          *reinterpret_cast<const float2*>(&sm.pr[pbase + j * K_NF + n]);
      S[j] = fmaf(w0, fabsf(fmaf(p2.x, Wm1, -t0)), S[j]);
      S[j] = fmaf(w1, fabsf(fmaf(p2.y, Wm1, -t1)), S[j]);
    }
  }

  const float tx = sm.gt[g * K_NF + 3];
  const float ty = sm.gt[g * K_NF + 2];
  const float tt = tx * tx + ty * ty;
  const float tth = sm.gt[g * K_NF + 4];
  const float mk = sm.maskf[g];

#pragma unroll
  for (int j = 0; j < 8; ++j) {
    const int pl = prow0 + j;
    const float d2 = sm.pp[pl] + tt - 2.f * crossv[j];
    const float dist = sqrtf(fmaxf(d2, 0.f) + 1e-8f);
    const float theta = fabsf(sm.pr[pl * K_NF + 4] - tth);
    const float ov = 30.f * VC - S[j];
    const float un = 30.f * VC + S[j];
    const float io = (ov / (un + 1e-9f)) * mk;      // padded GT -> 0
    const float tot = 4.f * sm.cls[pl] + 5.f * (dist + 2.f * theta)
                    + 2.f * (1.f - io) + 1e5f * (1.f - mk);
    const size_t o = ((size_t)(b * K_T + g)) * K_P + (size_t)(p0 + pl);
    cost[o] = tot;
    iou_out[o] = io;
  }
}

// ---------------------------------------------------------------------------
// Kernel B: per-(b,t) top-Q.  Q max-extractions over IoU -> dyn_k; Q
// min-extractions over cost (tie-break: lower index, matching lax.top_k).
// ---------------------------------------------------------------------------
__global__ __launch_bounds__(256) void topq_kernel(
    const float* __restrict__ cost, const float* __restrict__ iou,
    float* __restrict__ candC, int* __restrict__ candI,
    int* __restrict__ dynk) {
  __shared__ float v[K_P];
  __shared__ float rv[256];
  __shared__ int ri[256];
  __shared__ float s_sum;
  const int b = blockIdx.y, t = blockIdx.x, tid = threadIdx.x;
  const size_t row = ((size_t)(b * K_T + t)) * K_P;

  // ---- pass 1: IoU top-Q sum -> dyn_k ----
  for (int i = tid; i < K_P; i += 256) v[i] = iou[row + i];
  if (tid == 0) s_sum = 0.f;
  __syncthreads();
  for (int q = 0; q < K_Q; ++q) {
    float bv = -1e30f; int bi = K_P;
    for (int i = tid; i < K_P; i += 256) {
      const float x = v[i];
      if (x > bv || (x == bv && i < bi)) { bv = x; bi = i; }
    }
    rv[tid] = bv; ri[tid] = bi;
    __syncthreads();
    for (int s = 128; s > 0; s >>= 1) {
      if (tid < s) {
        const float ov = rv[tid + s]; const int oi = ri[tid + s];
        if (ov > rv[tid] || (ov == rv[tid] && oi < ri[tid])) {
          rv[tid] = ov; ri[tid] = oi;
        }
      }
      __syncthreads();
    }
    if (tid == 0) { s_sum += rv[0]; v[ri[0]] = -1e30f; }
    __syncthreads();
  }
  if (tid == 0) {
    int k = (int)s_sum;                      // astype(int32): trunc toward zero
    k = k < 1 ? 1 : (k > K_P ? K_P : k);
    dynk[b * K_T + t] = k;
  }
  __syncthreads();

  // ---- pass 2: Q lowest-cost candidates ----
  for (int i = tid; i < K_P; i += 256) v[i] = cost[row + i];
  __syncthreads();
  for (int q = 0; q < K_Q; ++q) {
    float bv = 1e30f; int bi = K_P;
    for (int i = tid; i < K_P; i += 256) {
      const float x = v[i];
      if (x < bv || (x == bv && i < bi)) { bv = x; bi = i; }
    }
    rv[tid] = bv; ri[tid] = bi;
    __syncthreads();
    for (int s = 128; s > 0; s >>= 1) {
      if (tid < s) {
        const float ov = rv[tid + s]; const int oi = ri[tid + s];
        if (ov < rv[tid] || (ov == rv[tid] && oi < ri[tid])) {
          rv[tid] = ov; ri[tid] = oi;
        }
      }
      __syncthreads();
    }
    if (tid == 0) {
      candC[(b * K_T + t) * K_Q + q] = rv[0];
      candI[(b * K_T + t) * K_Q + q] = ri[0];
      v[ri[0]] = 1e30f;
    }
    __syncthreads();
  }
}

// ---------------------------------------------------------------------------
// Kernel C: sequential scan over GTs (matches lax.scan order). Candidate
// indices within one GT step are distinct -> parallel update is race-free.
// ---------------------------------------------------------------------------
__global__ __launch_bounds__(256) void assign_kernel(
    const int* __restrict__ masks, const float* __restrict__ candC,
    const int* __restrict__ candI, const int* __restrict__ dynk,
    float* __restrict__ out) {
  __shared__ float minc[K_P];
  __shared__ int tg[K_P];
  const int b = blockIdx.x, tid = threadIdx.x;
  for (int i = tid; i < K_P; i += 256) { minc[i] = 1e8f; tg[i] = -1; }
  __syncthreads();
  for (int t = 0; t < K_T; ++t) {
    if (masks[b * K_T + t] != 0) {           // uniform branch per block
      if (tid < K_Q) {
        const int k = dynk[b * K_T + t];
        if (tid < k) {
          const int idx = candI[(b * K_T + t) * K_Q + tid];
          const float c = candC[(b * K_T + t) * K_Q + tid];
          if (c < minc[idx]) { minc[idx] = c; tg[idx] = t; }
        }
      }
    }
    __syncthreads();
  }
  const size_t BP = (size_t)K_B * K_P;
  for (int i = tid; i < K_P; i += 256) {
    out[(size_t)b * K_P + i] = (tg[i] >= 0) ? 1.f : 0.f;   // bool output
    out[BP + (size_t)b * K_P + i] = (float)tg[i];          // matched index
  }
}

// ---------------------------------------------------------------------------
extern "C" void kernel_launch(void* const* d_in, const int* in_sizes, int n_in,
                              void* d_out, int out_size, void* d_ws,
                              size_t ws_size, hipStream_t stream) {
  (void)in_sizes; (void)n_in; (void)out_size; (void)ws_size;
  const float* preds   = (const float*)d_in[0];   // [B,P,78]
  const float* targets = (const float*)d_in[1];   // [B,T,78]
  const int*   masks   = (const int*)d_in[2];     // [B,T]
  const int*   imgw    = (const int*)d_in[3];     // scalar on device
  // d_in[4]=img_h, d_in[5]=current_iter unused.

  float* wsF = (float*)d_ws;
  const size_t BTP = (size_t)K_B * K_T * K_P;     // 2,097,152
  const size_t BTQ = (size_t)K_B * K_T * K_Q;     // 5,120
  float* cost  = wsF;
  float* iouW  = wsF + BTP;
  float* candC = wsF + 2 * BTP;
  int*   candI = (int*)(candC + BTQ);
  int*   dynk  = candI + BTQ;

  dim3 gA(K_P / 16, K_B);
  cost_iou_kernel<<<gA, 128, 0, stream>>>(preds, targets, masks, imgw,
                                          cost, iouW);
  dim3 gB(K_T, K_B);
  topq_kernel<<<gB, 256, 0, stream>>>(cost, iouW, candC, candI, dynk);
  assign_kernel<<<K_B, 256, 0, stream>>>(masks, candC, candI, dynk,
                                         (float*)d_out);
}